// Model_90675349553219
// MI455X (gfx1250) — compile-verified
//
#include <hip/hip_runtime.h>

typedef __attribute__((ext_vector_type(2))) float v2f;
typedef __attribute__((ext_vector_type(8))) float v8f;

#define N_NODES 100000
#define DIM 128

// ---------------- degree kernels ----------------
__global__ void deg_count_kernel(const int* __restrict__ dst, float* __restrict__ deg, int E) {
    int e = blockIdx.x * blockDim.x + threadIdx.x;
    if (e < E) atomicAdd(&deg[dst[e]], 1.0f);
}

__global__ void inv_deg_kernel(float* __restrict__ deg, int n) {
    int i = blockIdx.x * blockDim.x + threadIdx.x;
    if (i < n) deg[i] = 1.0f / fmaxf(deg[i], 1.0f);
}

// ---------------- mean-aggregation scatter ----------------
// one wave per edge; lane owns a 4-float chunk of the 128-float feature row
__global__ void scatter_add_kernel(const float* __restrict__ h, const int* __restrict__ src,
                                   const int* __restrict__ dst, float* __restrict__ neigh, int E) {
    unsigned tid = blockIdx.x * blockDim.x + threadIdx.x;
    unsigned e = tid >> 5;
    if (e >= (unsigned)E) return;
    unsigned c = (tid & 31u) * 4u;
    int s = src[e], d = dst[e];
    const float4 v = *(const float4*)(h + (size_t)s * DIM + c);
    float* o = neigh + (size_t)d * DIM + c;
    atomicAdd(o + 0, v.x);
    atomicAdd(o + 1, v.y);
    atomicAdd(o + 2, v.z);
    atomicAdd(o + 3, v.w);
}

// ---------------- weight pre-swizzle into WMMA B-fragment order ----------------
// pack[((ks*NT + t)*32 + lane)*2 + {0,1}] = { w[(4ks+2hi+0)*ncols + col], w[(4ks+2hi+1)*ncols + col] }
// col = 16t + (lane&15) clamped; hi = lane>>4.  Lets the matmul load each B fragment
// as ONE global_load_b64 with a compile-time immediate offset.
template<int NT>
__global__ void pack_b_kernel(const float* __restrict__ w, float* __restrict__ pack, int ncols) {
    int idx = blockIdx.x * blockDim.x + threadIdx.x;   // 32 ksteps * NT tiles * 32 lanes
    if (idx >= 32 * NT * 32) return;
    int lane = idx & 31;
    int t    = (idx >> 5) % NT;
    int ks   = idx / (32 * NT);
    int lr = lane & 15, hi = lane >> 4;
    int col = t * 16 + lr;
    if (col >= ncols) col = ncols - 1;                 // clamped; masked at store time
    int kk = ks * 4 + 2 * hi;
    pack[(size_t)idx * 2 + 0] = w[(size_t)kk * ncols + col];
    pack[(size_t)idx * 2 + 1] = w[(size_t)(kk + 1) * ncols + col];
}

// ---------------- dual-input SAGE matmul via f32 WMMA ----------------
// out[m,n] = sum_k h[m,k]*ws[k,n] + (neigh[m,k]*invdeg[m])*wn[k,n] + bias[n] (+ReLU)
// One wave32 per 16-row strip; NT 16-col tiles. Weights pre-packed (see above).
template<int NT, bool RELU>
__global__ void __launch_bounds__(128)
sage_matmul_kernel(const float* __restrict__ h, const float* __restrict__ neigh,
                   const float* __restrict__ invdeg,
                   const float* __restrict__ pack_s, const float* __restrict__ pack_n,
                   const float* __restrict__ bias,
                   float* __restrict__ out, int ncols) {
    const int wave  = threadIdx.x >> 5;
    const int lane  = threadIdx.x & 31;
    const int strip = blockIdx.x * (blockDim.x >> 5) + wave;
    if (strip >= N_NODES / 16) return;   // uniform per wave -> whole wave exits together

    const int m0  = strip * 16;
    const int lr  = lane & 15;
    const int hi  = lane >> 4;
    const int rowA = m0 + lr;
    const float idg = invdeg[rowA];

    // A fragment: lane<16 holds {K,K+1} of row M=lr; lane>=16 holds {K+2,K+3}
    const float* hrow = h     + (size_t)rowA * DIM + 2 * hi;
    const float* nrow = neigh + (size_t)rowA * DIM + 2 * hi;
    const float* ps = pack_s + (size_t)lane * 2;   // per-lane base; rest is immediate offsets
    const float* pn = pack_n + (size_t)lane * 2;

    v8f acc[NT] = {};

    #pragma unroll
    for (int ks = 0; ks < 32; ++ks) {
        const int k = ks * 4;
        v2f a_s = *(const v2f*)(hrow + k);
        v2f a_n = *(const v2f*)(nrow + k);
        a_n.x *= idg;
        a_n.y *= idg;
        #pragma unroll
        for (int t = 0; t < NT; ++t) {
            v2f b_s = *(const v2f*)(ps + (ks * NT + t) * 64);
            v2f b_n = *(const v2f*)(pn + (ks * NT + t) * 64);
            acc[t] = __builtin_amdgcn_wmma_f32_16x16x4_f32(false, a_s, false, b_s,
                                                           (short)0, acc[t], false, false);
            acc[t] = __builtin_amdgcn_wmma_f32_16x16x4_f32(false, a_n, false, b_n,
                                                           (short)0, acc[t], false, false);
        }
    }

    // C/D layout: VGPR i -> row m0+i (lanes 0-15) / m0+i+8 (lanes 16-31), col = 16t + lr
    #pragma unroll
    for (int t = 0; t < NT; ++t) {
        int col = t * 16 + lr;
        if (col < ncols) {
            float bv = bias[col];
            #pragma unroll
            for (int i = 0; i < 8; ++i) {
                int row = m0 + i + hi * 8;
                float v = acc[t][i] + bv;
                if (RELU) v = fmaxf(v, 0.0f);
                out[(size_t)row * ncols + col] = v;
            }
        }
    }
}

// ---------------- launch ----------------
extern "C" void kernel_launch(void* const* d_in, const int* in_sizes, int n_in,
                              void* d_out, int out_size, void* d_ws, size_t ws_size,
                              hipStream_t stream) {
    const float* x           = (const float*)d_in[0];
    const int*   src         = (const int*)d_in[1];
    const int*   dst         = (const int*)d_in[2];
    const float* w_self      = (const float*)d_in[3];
    const float* w_neigh     = (const float*)d_in[4];
    const float* b           = (const float*)d_in[5];
    const float* w_self_out  = (const float*)d_in[6];
    const float* w_neigh_out = (const float*)d_in[7];
    const float* b_out       = (const float*)d_in[8];
    float* out = (float*)d_out;
    const int E = in_sizes[1];

    const size_t feat_elems = (size_t)N_NODES * DIM;
    const size_t pack8_elems = 32 * 8 * 32 * 2;   // 16384 floats per hidden weight matrix
    const size_t pack3_elems = 32 * 3 * 32 * 2;   // 6144 floats per final weight matrix

    float* hA     = (float*)d_ws;
    float* hB     = hA + feat_elems;
    float* neigh  = hB + feat_elems;
    float* invdeg = neigh + feat_elems;
    float* packs  = invdeg + N_NODES;             // 5*2 hidden packs then 2 final packs

    // ---- pre-swizzle all weights into fragment order (tiny, L2-resident) ----
    for (int i = 0; i < 5; ++i) {
        pack_b_kernel<8><<<(32 * 8 * 32 + 255) / 256, 256, 0, stream>>>(
            w_self + (size_t)i * DIM * DIM, packs + (2 * i + 0) * pack8_elems, DIM);
        pack_b_kernel<8><<<(32 * 8 * 32 + 255) / 256, 256, 0, stream>>>(
            w_neigh + (size_t)i * DIM * DIM, packs + (2 * i + 1) * pack8_elems, DIM);
    }
    float* pack_out_s = packs + 10 * pack8_elems;
    float* pack_out_n = pack_out_s + pack3_elems;
    pack_b_kernel<3><<<(32 * 3 * 32 + 255) / 256, 256, 0, stream>>>(w_self_out,  pack_out_s, 47);
    pack_b_kernel<3><<<(32 * 3 * 32 + 255) / 256, 256, 0, stream>>>(w_neigh_out, pack_out_n, 47);

    // ---- in-degree -> 1/max(deg,1) ----
    hipMemsetAsync(invdeg, 0, N_NODES * sizeof(float), stream);
    deg_count_kernel<<<(E + 255) / 256, 256, 0, stream>>>(dst, invdeg, E);
    inv_deg_kernel<<<(N_NODES + 255) / 256, 256, 0, stream>>>(invdeg, N_NODES);

    const unsigned scatter_blocks = (unsigned)(((size_t)E * 32 + 255) / 256);
    const int mm_blocks = (N_NODES / 16 + 3) / 4;          // 4 waves (128 threads) per block
    const size_t feat_bytes = feat_elems * sizeof(float);

    auto aggregate = [&](const float* hcur) {
        hipMemsetAsync(neigh, 0, feat_bytes, stream);
        scatter_add_kernel<<<scatter_blocks, 256, 0, stream>>>(hcur, src, dst, neigh, E);
    };

    // layer 0: no ReLU
    aggregate(x);
    sage_matmul_kernel<8, false><<<mm_blocks, 128, 0, stream>>>(
        x, neigh, invdeg, packs + 0 * pack8_elems, packs + 1 * pack8_elems, b, hA, DIM);

    // layers 1..4: SAGE + ReLU, ping-pong hA/hB
    const float* cur = hA;
    float* nxt = hB;
    for (int i = 1; i < 5; ++i) {
        aggregate(cur);
        sage_matmul_kernel<8, true><<<mm_blocks, 128, 0, stream>>>(
            cur, neigh, invdeg,
            packs + (2 * i + 0) * pack8_elems, packs + (2 * i + 1) * pack8_elems,
            b + (size_t)i * DIM, nxt, DIM);
        float* tmp = (float*)cur; cur = nxt; nxt = tmp;
    }

    // final layer: 128 -> 47, no activation
    aggregate(cur);
    sage_matmul_kernel<3, false><<<mm_blocks, 128, 0, stream>>>(
        cur, neigh, invdeg, pack_out_s, pack_out_n, b_out, out, 47);
}